// BoundaryPredictor2_7816840479224
// MI455X (gfx1250) — compile-verified
//
#include <hip/hip_runtime.h>
#include <math.h>

// Problem constants (from reference: B=8, L=2048, D=1024)
#define DDIM  1024
#define LSEQ  2048
#define BATCH 8
#define NROWS (BATCH * LSEQ)          // 16384
#define EPSF  1.1920929e-07f
#define PRIORF 0.2f

typedef float v2f __attribute__((ext_vector_type(2)));
typedef float v8f __attribute__((ext_vector_type(8)));

// ---------------------------------------------------------------------------
// K1: per-row 1/max(||h||,1e-12).  One wave per row, shuffle reduction.
// ---------------------------------------------------------------------------
__global__ void k_rownorm(const float* __restrict__ hidden, float* __restrict__ rinv) {
    int wave = threadIdx.x >> 5;
    int lane = threadIdx.x & 31;
    int row  = blockIdx.x * 8 + wave;
    const float* p = hidden + (size_t)row * DDIM;
    float s = 0.f;
    for (int c = lane; c < DDIM; c += 32) { float v = p[c]; s += v * v; }
    s += __shfl_xor(s, 1);  s += __shfl_xor(s, 2);  s += __shfl_xor(s, 4);
    s += __shfl_xor(s, 8);  s += __shfl_xor(s, 16);
    if (lane == 0) rinv[row] = 1.f / fmaxf(sqrtf(s), 1e-12f);
}

// ---------------------------------------------------------------------------
// K2: M = Wq^T * Wk  (1024x1024x1024 fp32 GEMM via V_WMMA_F32_16X16X4_F32)
// One 16x16 tile of M per wave; Wq/Wk (4MB each) stay hot in the 192MB L2.
// Only ~6% of total FLOPs -> simple direct-load form is fine.
// ---------------------------------------------------------------------------
__global__ void k_gemm_M(const float* __restrict__ Wq, const float* __restrict__ Wk,
                         float* __restrict__ Mm) {
    int wave = threadIdx.x >> 5;
    int lane = threadIdx.x & 31;
    int half = lane >> 4, ln = lane & 15;
    int t  = blockIdx.x * 8 + wave;          // 4096 tiles total
    int ti = (t >> 6) * 16;                  // row tile
    int tj = (t & 63) * 16;                  // col tile
    v8f acc = {};
    for (int k0 = 0; k0 < DDIM; k0 += 4) {
        int kr = k0 + half * 2;              // A/B VGPR0 holds K={0,2}, VGPR1 K={1,3}
        v2f a, b;
        a.x = Wq[(size_t)kr * DDIM + ti + ln];
        a.y = Wq[(size_t)(kr + 1) * DDIM + ti + ln];
        b.x = Wk[(size_t)kr * DDIM + tj + ln];
        b.y = Wk[(size_t)(kr + 1) * DDIM + tj + ln];
        acc = __builtin_amdgcn_wmma_f32_16x16x4_f32(false, a, false, b,
                                                    (short)0, acc, false, false);
    }
#pragma unroll
    for (int v = 0; v < 8; ++v)              // C/D: row = v + 8*half, col = ln
        Mm[(size_t)(ti + v + half * 8) * DDIM + tj + ln] = acc[v];
}

// ---------------------------------------------------------------------------
// K3: t = (hidden * rinv) @ M, fused with the adjacent-row dot product.
// Block = 256 threads = 8 waves covers a 64-row x 128-col tile of t.
// Each wave owns one 16-col j-slice and 4 stacked row-tiles (32 acc VGPRs),
// so each B fragment feeds 4 WMMAs (4x less global B traffic than 1-tile).
// A chunk (64x64) staged in LDS, pitch 68 -> conflict-free ds_load_b64 of
// each lane's K-pair.  Epilogue writes fixed-order per-jtile partial dots
// with nh[r+1] (deterministic; reduced in K4 -> no atomics).
// ---------------------------------------------------------------------------
__global__ void k_gemm_T(const float* __restrict__ hidden, const float* __restrict__ rinv,
                         const float* __restrict__ Mm, float* __restrict__ partial) {
    __shared__ float As[64][68];             // 17 KB
    __shared__ float rs[64];
    int wave = threadIdx.x >> 5;
    int lane = threadIdx.x & 31;
    int half = lane >> 4, ln = lane & 15;
    int rb = blockIdx.x >> 3;                // 256 row blocks of 64
    int jb = blockIdx.x & 7;                 // 8 col blocks of 128
    int r0 = rb * 64;
    int j0 = jb * 128 + wave * 16;

    if (threadIdx.x < 64) rs[threadIdx.x] = rinv[r0 + threadIdx.x];
    __syncthreads();

    v8f acc[4] = {};
    for (int kc = 0; kc < DDIM; kc += 64) {
        // prefetch next K-chunk of B into cache (global_prefetch_b8)
        if (kc + 64 < DDIM)
            __builtin_prefetch(&Mm[(size_t)(kc + 64) * DDIM + j0 + ln], 0, 3);
#pragma unroll
        for (int i = 0; i < 16; ++i) {       // 4096 floats, coalesced 64/row
            int idx = threadIdx.x + i * 256;
            int row = idx >> 6, col = idx & 63;
            As[row][col] = hidden[(size_t)(r0 + row) * DDIM + kc + col] * rs[row];
        }
        __syncthreads();
        for (int kk = 0; kk < 64; kk += 4) {
            int kl = kk + half * 2;          // even -> 8B-aligned LDS pair
            int kg = kc + kl;
            v2f b;
            b.x = Mm[(size_t)kg * DDIM + j0 + ln];
            b.y = Mm[(size_t)(kg + 1) * DDIM + j0 + ln];
#pragma unroll
            for (int mt = 0; mt < 4; ++mt) {
                v2f a = *(const v2f*)&As[mt * 16 + ln][kl];
                acc[mt] = __builtin_amdgcn_wmma_f32_16x16x4_f32(
                    false, a, false, b, (short)0, acc[mt], false, false);
            }
        }
        __syncthreads();
    }

#pragma unroll
    for (int mt = 0; mt < 4; ++mt) {
#pragma unroll
        for (int v = 0; v < 8; ++v) {
            int r = r0 + mt * 16 + v + half * 8;
            int l = r & (LSEQ - 1);
            float p = 0.f;
            if (l != LSEQ - 1)               // last token of a batch has no pair
                p = acc[mt][v] * (hidden[(size_t)(r + 1) * DDIM + j0 + ln] * rinv[r + 1]);
            // reduce across the 16 lanes of this half-wave (xor<=8 stays in half)
            p += __shfl_xor(p, 1);  p += __shfl_xor(p, 2);
            p += __shfl_xor(p, 4);  p += __shfl_xor(p, 8);
            if (l != LSEQ - 1 && ln == 0)
                partial[(size_t)r * 64 + (j0 >> 4)] = p;
        }
    }
}

// ---------------------------------------------------------------------------
// K4: cos -> probs -> RelaxedBernoulli rsample -> straight-through hard bits.
// ---------------------------------------------------------------------------
__global__ void k_boundary(const float* __restrict__ partial, const float* __restrict__ noise,
                           const int* __restrict__ mask, int* __restrict__ hb,
                           float* __restrict__ hardArr) {
    int gid = blockIdx.x * 256 + threadIdx.x;
    if (gid >= NROWS) return;
    int l = gid & (LSEQ - 1);
    float probs;
    if (l == 0) {
        probs = 1.f;                         // padded cos = -1
    } else {
        const float* pp = partial + (size_t)(gid - 1) * 64;
        float cosv = 0.f;
        for (int t = 0; t < 64; ++t) cosv += pp[t];   // deterministic order
        probs = fminf(fmaxf((1.f - cosv) * 0.5f, 0.f), 1.f);
    }
    float p      = fminf(fmaxf(probs, EPSF), 1.f - EPSF);
    float logits = logf(p) - log1pf(-p);
    float u      = fminf(fmaxf(noise[gid], EPSF), 1.f - EPSF);
    float lgst   = logf(u) - log1pf(-u);
    float soft   = 1.f / (1.f + expf(-(logits + lgst)));   // TEMP = 1
    float mf     = (float)mask[gid];
    float hard   = (soft > 0.5f ? 1.f : 0.f) * mf;
    hardArr[gid] = hard;
    hb[gid]      = (hard > 0.5f) ? 1 : 0;
}

// ---------------------------------------------------------------------------
// K5: per-batch serial scan: segment starts/counts, pooled mask, counters.
// ---------------------------------------------------------------------------
__global__ void k_segments(const int* __restrict__ hb, const float* __restrict__ hardArr,
                           const int* __restrict__ mask, int* __restrict__ segCnt,
                           int* __restrict__ segStart, float* __restrict__ pooledMask,
                           float* __restrict__ nbArr, float* __restrict__ totArr) {
    int b = threadIdx.x;
    if (b >= BATCH) return;
    int base = b * LSEQ;
    for (int s = 0; s < LSEQ; ++s) { segCnt[base + s] = 0; pooledMask[base + s] = 0.f; }
    int cum = 0; float nb = 0.f, tot = 0.f;
    for (int l = 0; l < LSEQ; ++l) {
        int h = hb[base + l];
        if (segCnt[base + cum] == 0) segStart[base + cum] = l;
        segCnt[base + cum] += 1;
        tot += (mask[base + l] != 0) ? 1.f : 0.f;
        nb  += hardArr[base + l];
        if (h) { pooledMask[base + cum] = (float)mask[base + l]; ++cum; }
    }
    nbArr[b] = nb;  totArr[b] = tot;
}

// ---------------------------------------------------------------------------
// K6: binomial-prior loss (scalars).
// ---------------------------------------------------------------------------
__global__ void k_loss(const float* __restrict__ nbArr, const float* __restrict__ totArr,
                       float* __restrict__ outScalars) {
    if (threadIdx.x != 0 || blockIdx.x != 0) return;
    float kk = 0.f, n = 0.f;
    for (int b = 0; b < BATCH; ++b) { kk += nbArr[b]; n += totArr[b]; }
    float log_pmf = lgammaf(n + 1.f) - lgammaf(kk + 1.f) - lgammaf(n - kk + 1.f)
                  + kk * logf(PRIORF) + (n - kk) * log1pf(-PRIORF);
    outScalars[0] = -log_pmf / n;
    outScalars[1] = kk;
    outScalars[2] = n;
}

// ---------------------------------------------------------------------------
// K7: segment mean pooling.  One block per (b, segment), coalesced over D.
// Empty segments naturally produce 0 via sums/(cnt+1e-9).
// ---------------------------------------------------------------------------
__global__ void k_pool(const float* __restrict__ hidden, const int* __restrict__ segCnt,
                       const int* __restrict__ segStart, float* __restrict__ pooled) {
    int bs   = blockIdx.x;                  // b*LSEQ + s
    int cnt  = segCnt[bs];
    int b    = bs >> 11;
    int start = (cnt > 0) ? segStart[bs] : 0;
    size_t rowBase = ((size_t)(b * LSEQ + start)) * DDIM;
    float inv = 1.f / ((float)cnt + 1e-9f);
    for (int d = threadIdx.x; d < DDIM; d += 256) {
        float s = 0.f;
        for (int l = 0; l < cnt; ++l) s += hidden[rowBase + (size_t)l * DDIM + d];
        pooled[(size_t)bs * DDIM + d] = s * inv;
    }
}

// ---------------------------------------------------------------------------
extern "C" void kernel_launch(void* const* d_in, const int* in_sizes, int n_in,
                              void* d_out, int out_size, void* d_ws, size_t ws_size,
                              hipStream_t stream) {
    (void)in_sizes; (void)n_in; (void)out_size; (void)ws_size;
    const float* hidden = (const float*)d_in[0];   // [8,2048,1024] f32
    const int*   mask   = (const int*)  d_in[1];   // [8,2048] i32
    const float* noise  = (const float*)d_in[2];   // [8,2048] f32
    const float* Wq     = (const float*)d_in[3];   // [1024,1024] f32
    const float* Wk     = (const float*)d_in[4];   // [1024,1024] f32

    // Workspace layout (~8.7 MB total)
    char* ws = (char*)d_ws;
    size_t off = 0;
    float* Mm       = (float*)(ws + off); off += (size_t)DDIM * DDIM * 4;   // 4 MiB
    float* rinv     = (float*)(ws + off); off += (size_t)NROWS * 4;
    float* partial  = (float*)(ws + off); off += (size_t)NROWS * 64 * 4;    // 4 MiB
    int*   hb       = (int*)  (ws + off); off += (size_t)NROWS * 4;
    float* hardArr  = (float*)(ws + off); off += (size_t)NROWS * 4;
    int*   segCnt   = (int*)  (ws + off); off += (size_t)NROWS * 4;
    int*   segStart = (int*)  (ws + off); off += (size_t)NROWS * 4;
    float* nbArr    = (float*)(ws + off); off += 64;
    float* totArr   = (float*)(ws + off); off += 64;

    // Output layout: pooled | pooled_mask | loss | num_boundaries | total_positions
    float* pooled     = (float*)d_out;
    float* pooledMask = pooled + (size_t)NROWS * DDIM;
    float* scalars    = pooledMask + NROWS;

    k_rownorm <<<NROWS / 8,          256, 0, stream>>>(hidden, rinv);
    k_gemm_M  <<<(64 * 64) / 8,      256, 0, stream>>>(Wq, Wk, Mm);
    k_gemm_T  <<<(NROWS / 64) * 8,   256, 0, stream>>>(hidden, rinv, Mm, partial);
    k_boundary<<<NROWS / 256,        256, 0, stream>>>(partial, noise, mask, hb, hardArr);
    k_segments<<<1,                   32, 0, stream>>>(hb, hardArr, mask, segCnt, segStart,
                                                       pooledMask, nbArr, totArr);
    k_loss    <<<1,                   32, 0, stream>>>(nbArr, totArr, scalars);
    k_pool    <<<NROWS,              256, 0, stream>>>(hidden, segCnt, segStart, pooled);
}